// Consis_Reg_13709535609589
// MI455X (gfx1250) — compile-verified
//
#include <hip/hip_runtime.h>
#include <hip/hip_bf16.h>

// ---------------------------------------------------------------------------
// Consis_Reg (MSE branch), algebraically reduced:
//   loss = sum_c [ 2*sum_{i in c}||A_i||^2 - (2/n_c) * || sum_{i in c} A_i ||^2 ]
// One-hot fp32 GEMM (M = L^T A, M2 = L^T (A*A)) on V_WMMA_F32_16X16X4_F32.
// The 16 MB input is streamed exactly once via GLOBAL_LOAD_ASYNC_TO_LDS_B128
// (ASYNCcnt) through an 8-deep LDS ring with prefetch distance 7 (~3.5 KB in
// flight per wave). Loops are kept rolled and the kernel is pinned to
// 1 WG/EU so the 16 v8f WMMA accumulators stay in registers (no spills).
// ---------------------------------------------------------------------------

typedef float v2f __attribute__((ext_vector_type(2)));
typedef float v8f __attribute__((ext_vector_type(8)));

#define N_ROWS      8192
#define DIM         512
#define NCLS        64
#define NWG         64                      // row-chunk workgroups
#define ROWS_PER_WG (N_ROWS / NWG)          // 128
#define KSTEPS      (ROWS_PER_WG / 4)       // 32 (K=4 per WMMA)
#define WAVES_K1    16                      // 512 threads; each wave owns 32 dims
#define NBUF        8                       // LDS ring depth (power of 2)
#define PF          7                       // prefetch distance (< NBUF: no reuse race)

// d_ws byte offsets
#define WS_COUNTS    0                      // 64 ints
#define WS_BLOCKSUMS 256                    // 128 floats
#define WS_WAVEM2    4096                   // NWG*16*64 floats   = 256 KB
#define WS_PARTIALM  (1 << 19)              // NWG*64*512 floats  = 8 MB

// ---- CDNA5 async global->LDS helpers (ASYNCcnt) ---------------------------
__device__ __forceinline__ void async_load_b128(unsigned lds_off, const float* gaddr) {
    // vdst slot carries the LDS byte address; GV mode (64-bit VGPR address).
    asm volatile("global_load_async_to_lds_b128 %0, %1, off"
                 :: "v"(lds_off), "v"(gaddr)
                 : "memory");
}
__device__ __forceinline__ void wait_async_const(int n) {
    asm volatile("s_wait_asynccnt %0" :: "i"(n) : "memory");
}
__device__ __forceinline__ void wait_ds0() {
    asm volatile("s_wait_dscnt 0x0" ::: "memory");
}

__global__ __launch_bounds__(64)
void cr_init_counts(int* __restrict__ counts) {
    counts[threadIdx.x] = 0;
}

__global__ __launch_bounds__(256)
void cr_count(const int* __restrict__ tgt, int* __restrict__ counts) {
    int i = blockIdx.x * 256 + threadIdx.x;
    if (i < N_ROWS) atomicAdd(&counts[tgt[i]], 1);   // integer atomics: deterministic
}

// Main pass: per-WG partial class sums (M) and class sums-of-squares (M2)
// via fp32 WMMA with a one-hot A-operand. EXEC is all-ones around the WMMAs.
__global__ __launch_bounds__(512, 1)
void cr_partials(const float* __restrict__ A, const int* __restrict__ tgt,
                 float* __restrict__ partialM, float* __restrict__ waveM2) {
    // LDS ring: [wave][buf][4 rows][32 dims] floats = 16*8*512B = 64 KB
    __shared__ __align__(16) float sbuf[WAVES_K1 * NBUF * 128];

    const int wg   = blockIdx.x;
    const int tid  = threadIdx.x;
    const int w    = tid >> 5;        // wave id 0..15 -> dims [w*32, w*32+32)
    const int lane = tid & 31;
    const int sub  = lane & 15;
    const int half = lane >> 4;       // lane half selects K/M per ISA layouts
    const int dimbase = w * 32;
    const int rowbase = wg * ROWS_PER_WG;
    const int kx   = 2 * half;        // 16x16x4 f32: V0 = K0|K2, V1 = K1|K3

    // Async copy geometry: lane L fetches 16 B of row (rb + L/8),
    // dims [dimbase + (L%8)*4, +4) -> LDS offset L*16 within the step buffer.
    const int lrow = lane >> 3;
    const int lcol = lane & 7;
    const float* gbase = A + (size_t)(rowbase + lrow) * DIM + dimbase + lcol * 4;
    float* wbuf = &sbuf[w * NBUF * 128];
    const unsigned lds_base = (unsigned)(size_t)wbuf + (unsigned)(lane * 16);

    v8f acc [4][2];                   // M  tiles: 4 class-tiles x 2 dim-tiles
    v8f acc2[4][2];                   // M2 tiles
#pragma unroll
    for (int mt = 0; mt < 4; ++mt)
#pragma unroll
        for (int nt = 0; nt < 2; ++nt) { acc[mt][nt] = {}; acc2[mt][nt] = {}; }

    // Prologue: fill the pipeline PF deep.
#pragma unroll
    for (int p = 0; p < PF; ++p)
        async_load_b128(lds_base + p * 512, gbase + (size_t)p * 4 * DIM);

    auto step_body = [&](int s) {
        const int rb = rowbase + s * 4;
        const int t0 = tgt[rb + kx];
        const int t1 = tgt[rb + kx + 1];
        const float* cur = wbuf + (s & (NBUF - 1)) * 128;  // [4][32] tile slice

        v2f bf[2], bq[2];
#pragma unroll
        for (int nt = 0; nt < 2; ++nt) {
            const float x = cur[kx * 32 + nt * 16 + sub];        // row K=kx
            const float y = cur[(kx + 1) * 32 + nt * 16 + sub];  // row K=kx+1
            bf[nt] = (v2f){x, y};
            bq[nt] = (v2f){x * x, y * y};
        }
#pragma unroll
        for (int mt = 0; mt < 4; ++mt) {
            const int m = mt * 16 + sub;
            v2f af = (v2f){ (t0 == m) ? 1.0f : 0.0f,
                            (t1 == m) ? 1.0f : 0.0f };
#pragma unroll
            for (int nt = 0; nt < 2; ++nt) {
                acc[mt][nt]  = __builtin_amdgcn_wmma_f32_16x16x4_f32(
                    false, af, false, bf[nt], (short)0, acc[mt][nt],  false, false);
                acc2[mt][nt] = __builtin_amdgcn_wmma_f32_16x16x4_f32(
                    false, af, false, bq[nt], (short)0, acc2[mt][nt], false, false);
            }
        }
    };

    // Steady state (rolled; keeps exactly one copy of the step body live):
    // ds-fence before ring-slot reuse (ASYNCcnt and DScnt are unordered),
    // issue the refill, then wait until step s's copy (oldest of <=8) landed.
#pragma unroll 1
    for (int s = 0; s < KSTEPS - PF; ++s) {
        wait_ds0();
        async_load_b128(lds_base + ((s + PF) & (NBUF - 1)) * 512,
                        gbase + (size_t)(s + PF) * 4 * DIM);
        wait_async_const(PF);
        step_body(s);
    }
    // Tail (rolled): nothing left to issue; first iteration drains the ring
    // (youngest load was issued one iteration ago), the rest wait for free.
#pragma unroll 1
    for (int s = KSTEPS - PF; s < KSTEPS; ++s) {
        wait_async_const(0);
        step_body(s);
    }

    // Write per-WG partial M: C/D layout -> VGPR r holds M=r (lanes 0-15), M=r+8 (16-31)
    float* pM = partialM + (size_t)wg * (NCLS * DIM);
#pragma unroll
    for (int mt = 0; mt < 4; ++mt)
#pragma unroll
        for (int nt = 0; nt < 2; ++nt) {
            const int d = dimbase + nt * 16 + sub;
#pragma unroll
            for (int r = 0; r < 8; ++r) {
                const int cls = mt * 16 + r + half * 8;
                pM[cls * DIM + d] = acc[mt][nt][r];
            }
        }

    // M2: reduce this wave's 32 dims (xor-shuffles within 16-lane halves),
    // one scalar per class per wave -> deterministic second stage.
#pragma unroll
    for (int mt = 0; mt < 4; ++mt) {
        v8f s = acc2[mt][0] + acc2[mt][1];
#pragma unroll
        for (int r = 0; r < 8; ++r) {
            float v = s[r];
            v += __shfl_xor(v, 1, 32);
            v += __shfl_xor(v, 2, 32);
            v += __shfl_xor(v, 4, 32);
            v += __shfl_xor(v, 8, 32);
            if (sub == 0) {
                const int cls = mt * 16 + r + half * 8;
                waveM2[(wg * WAVES_K1 + w) * NCLS + cls] = v;
            }
        }
    }
}

// Stage 1 reduce: 32768 threads, one per (c,d). Fixed-order loops -> deterministic.
__global__ __launch_bounds__(256)
void cr_reduce1(const float* __restrict__ partialM, const float* __restrict__ waveM2,
                const int* __restrict__ counts, float* __restrict__ blockSums) {
    __shared__ float sm[256];
    const int tg = blockIdx.x * 256 + threadIdx.x;   // 0..32767 ; c = tg/512
    const int c  = tg >> 9;

    float m = 0.0f;
#pragma unroll 8
    for (int wg = 0; wg < NWG; ++wg)
        m += partialM[(size_t)wg * (NCLS * DIM) + tg];

    // waveM2 has NWG*16*64 = 65536 entries; fold two per thread (sum over all).
    const float m2 = waveM2[tg] + waveM2[tg + 32768];
    const float nc = (float)counts[c];
    sm[threadIdx.x] = 2.0f * m2 - 2.0f * m * m / nc;
    __syncthreads();
#pragma unroll
    for (int s = 128; s > 0; s >>= 1) {
        if (threadIdx.x < s) sm[threadIdx.x] += sm[threadIdx.x + s];
        __syncthreads();
    }
    if (threadIdx.x == 0) blockSums[blockIdx.x] = sm[0];
}

// Stage 2 reduce: single block, fixed tree order -> bitwise deterministic output.
__global__ __launch_bounds__(128)
void cr_reduce2(const float* __restrict__ blockSums, float* __restrict__ out) {
    __shared__ float sm[128];
    sm[threadIdx.x] = blockSums[threadIdx.x];
    __syncthreads();
#pragma unroll
    for (int s = 64; s > 0; s >>= 1) {
        if (threadIdx.x < s) sm[threadIdx.x] += sm[threadIdx.x + s];
        __syncthreads();
    }
    if (threadIdx.x == 0) out[0] = sm[0];
}

extern "C" void kernel_launch(void* const* d_in, const int* in_sizes, int n_in,
                              void* d_out, int out_size, void* d_ws, size_t ws_size,
                              hipStream_t stream) {
    const float* A   = (const float*)d_in[0];
    const int*   tgt = (const int*)d_in[1];
    float*       out = (float*)d_out;

    char* ws = (char*)d_ws;
    int*   counts    = (int*)(ws + WS_COUNTS);
    float* blockSums = (float*)(ws + WS_BLOCKSUMS);
    float* waveM2    = (float*)(ws + WS_WAVEM2);
    float* partialM  = (float*)(ws + WS_PARTIALM);

    cr_init_counts<<<1, 64, 0, stream>>>(counts);
    cr_count<<<N_ROWS / 256, 256, 0, stream>>>(tgt, counts);
    cr_partials<<<NWG, 512, 0, stream>>>(A, tgt, partialM, waveM2);
    cr_reduce1<<<(NCLS * DIM) / 256, 256, 0, stream>>>(partialM, waveM2, counts, blockSums);
    cr_reduce2<<<1, 128, 0, stream>>>(blockSums, out);
}